// ReadoutLayerMultidim_67302137528973
// MI455X (gfx1250) — compile-verified
//
#include <hip/hip_runtime.h>
#include <hip/hip_bf16.h>

typedef __attribute__((ext_vector_type(2))) float v2f;
typedef __attribute__((ext_vector_type(8))) float v8f;

// ---------------------------------------------------------------------------
// Kernel 1: base[b,h] = relu( mean_n(x[b,n,:]) @ W_pool )[f] @ W1[0:128] + b1
// One block per batch element. Tiny.
// ---------------------------------------------------------------------------
__global__ __launch_bounds__(256) void base_kernel(const float* __restrict__ X,
                                                   const float* __restrict__ Wp,
                                                   const float* __restrict__ W1,
                                                   const float* __restrict__ b1,
                                                   float* __restrict__ base) {
  __shared__ float smean[128];
  __shared__ float srp[128];
  const int b = blockIdx.x;
  const int t = threadIdx.x;
  if (t < 128) {
    float s = 0.f;
    const float* xb = X + b * 256 * 128 + t;
    for (int n = 0; n < 256; ++n) s += xb[n * 128];
    smean[t] = s * (1.0f / 256.0f);
  }
  __syncthreads();
  if (t < 128) {
    float s = 0.f;
    for (int g = 0; g < 128; ++g) s += smean[g] * Wp[g * 128 + t];
    srp[t] = s > 0.f ? s : 0.f;  // relu(h_pooled)
  }
  __syncthreads();
  {
    float s = b1[t];
    for (int f = 0; f < 128; ++f) s += srp[f] * W1[f * 256 + t];
    base[b * 256 + t] = s;
  }
}

// ---------------------------------------------------------------------------
// Kernel 2: U[row, 0:256]   = relu(X[row]) @ W1[128:256] + base[row/256]
//           U[row, 256:512] = relu(X[row]) @ W1[256:384]
// rows = B*N = 1024, K = 128. fp32 WMMA: one wave per 16x16 tile, 32 k-steps.
// A frag (16x4 f32): lane l<16 holds A[l][k], A[l][k+1]; lanes 16-31 K=+2.
// B frag (4x16 f32): lane l<16 holds B[k][l], B[k+1][l]; lanes 16-31 K=+2.
// D (16x16 f32):     vgpr v, lanes 0-15 -> M=v, lanes 16-31 -> M=v+8; N=l%16.
// ---------------------------------------------------------------------------
__global__ __launch_bounds__(256) void gemm_u_kernel(const float* __restrict__ X,
                                                     const float* __restrict__ W1,
                                                     const float* __restrict__ base,
                                                     float* __restrict__ U) {
  const int lane = threadIdx.x & 31;
  const int wave = threadIdx.x >> 5;
  const int half = lane >> 4;   // selects K pair {0,1} vs {2,3}
  const int lmod = lane & 15;

  const int m0   = blockIdx.x * 16;                 // row tile (b*n)
  const int coln = blockIdx.y * 128 + wave * 16 + lmod;  // global col 0..511

  // A: relu(X[m0+lmod, k + 2*half .. +1])
  const float* arow = X + (m0 + lmod) * 128 + 2 * half;
  // B: W1 rows (128 + [coln>=256]*128 + k + 2*half), col (coln & 255)
  const int wrow0 = 128 + ((coln >= 256) ? 128 : 0) + 2 * half;
  const float* bcol = W1 + wrow0 * 256 + (coln & 255);

  v8f acc = {};
  for (int k = 0; k < 128; k += 4) {
    v2f a, bb;
    float ax = arow[k];
    float ay = arow[k + 1];
    a.x = ax > 0.f ? ax : 0.f;
    a.y = ay > 0.f ? ay : 0.f;
    bb.x = bcol[k * 256];
    bb.y = bcol[(k + 1) * 256];
    acc = __builtin_amdgcn_wmma_f32_16x16x4_f32(
        /*neg_a=*/false, a, /*neg_b=*/false, bb,
        /*c_mod=*/(short)0, acc, /*reuse_a=*/false, /*reuse_b=*/false);
  }

  const int bidx = m0 >> 8;  // batch index (16 | 256, so tile is batch-uniform)
  const float basev = (coln < 256) ? base[bidx * 256 + coln] : 0.f;
#pragma unroll
  for (int v = 0; v < 8; ++v) {
    const int row = m0 + v + 8 * half;
    U[row * 512 + coln] = acc[v] + basev;
  }
}

// ---------------------------------------------------------------------------
// Kernel 3: out[b,i,j] = sum_h relu(Si[b,i,h] + Uj[b,j,h]) * W2[h] + b2
// Tile 32 i x 32 j per block; Si broadcast per wave, UjT lane-consecutive.
// ---------------------------------------------------------------------------
__global__ __launch_bounds__(256) void out_kernel(const float* __restrict__ U,
                                                  const float* __restrict__ W2,
                                                  const float* __restrict__ b2,
                                                  float* __restrict__ out) {
  __shared__ float4 sSi[32][64];  // [i][hblk]  : U cols 0..255 (base folded)
  __shared__ float4 sUj[64][32];  // [hblk][j]  : U cols 256..511
  const int b  = blockIdx.z;
  const int i0 = blockIdx.y * 32;
  const int j0 = blockIdx.x * 32;
  const int t  = threadIdx.x;

  // Si tile: hb fastest -> coalesced global, per-wave rows
  for (int idx = t; idx < 2048; idx += 256) {
    const int hb = idx & 63, r = idx >> 6;
    sSi[r][hb] = ((const float4*)(U + (b * 256 + i0 + r) * 512))[hb];
  }
  // Uj tile (transposed in LDS): r fastest -> conflict-free LDS writes
  for (int idx = t; idx < 2048; idx += 256) {
    const int r = idx & 31, hb = idx >> 5;
    sUj[hb][r] = ((const float4*)(U + (b * 256 + j0 + r) * 512 + 256))[hb];
  }
  __syncthreads();

  const int lane = t & 31;  // j within tile
  const int iw   = t >> 5;  // wave id -> i base (broadcast LDS reads per wave)
  const float4* W24 = (const float4*)W2;  // uniform address -> scalar loads

  float acc0 = 0.f, acc1 = 0.f, acc2 = 0.f, acc3 = 0.f;
  for (int hb = 0; hb < 64; ++hb) {
    const float4 uj = sUj[hb][lane];
    const float4 w  = W24[hb];
#pragma unroll
    for (int ii = 0; ii < 4; ++ii) {
      const float4 si = sSi[iw + 8 * ii][hb];
      float s0 = si.x + uj.x; s0 = s0 > 0.f ? s0 : 0.f;
      float s1 = si.y + uj.y; s1 = s1 > 0.f ? s1 : 0.f;
      float s2 = si.z + uj.z; s2 = s2 > 0.f ? s2 : 0.f;
      float s3 = si.w + uj.w; s3 = s3 > 0.f ? s3 : 0.f;
      const float d = s0 * w.x + s1 * w.y + s2 * w.z + s3 * w.w;
      if (ii == 0) acc0 += d; else if (ii == 1) acc1 += d;
      else if (ii == 2) acc2 += d; else acc3 += d;
    }
  }
  const float bias = b2[0];
  const float accs[4] = {acc0, acc1, acc2, acc3};
#pragma unroll
  for (int ii = 0; ii < 4; ++ii) {
    const int i = i0 + iw + 8 * ii;
    out[(b * 256 + i) * 256 + j0 + lane] = accs[ii] + bias;
  }
}

// ---------------------------------------------------------------------------
extern "C" void kernel_launch(void* const* d_in, const int* in_sizes, int n_in,
                              void* d_out, int out_size, void* d_ws, size_t ws_size,
                              hipStream_t stream) {
  const float* X  = (const float*)d_in[0];  // node_embeddings [4,256,128]
  const float* Wp = (const float*)d_in[1];  // W_pool [128,128]
  const float* W1 = (const float*)d_in[2];  // W1 [384,256]
  const float* b1 = (const float*)d_in[3];  // b1 [256]
  const float* W2 = (const float*)d_in[4];  // W2 [256,1]
  const float* b2 = (const float*)d_in[5];  // b2 [1]
  float* out = (float*)d_out;               // [4,256,256,1]

  float* U    = (float*)d_ws;       // [1024][512]: cols 0..255 = Ui+base, 256..511 = Uj
  float* base = U + 1024 * 512;     // [4][256]

  base_kernel<<<4, 256, 0, stream>>>(X, Wp, W1, b1, base);
  gemm_u_kernel<<<dim3(64, 4), 256, 0, stream>>>(X, W1, base, U);
  out_kernel<<<dim3(8, 8, 4), 256, 0, stream>>>(U, W2, b2, out);
}